// HausdorffDistanceLoss_40458591928853
// MI455X (gfx1250) — compile-verified
//
#include <hip/hip_runtime.h>
#include <hip/hip_bf16.h>
#include <math.h>

typedef __attribute__((ext_vector_type(2))) float v2f;
typedef __attribute__((ext_vector_type(8))) float v8f;

#define BATCH 4
#define DD 32
#define HH 32
#define WW 16
#define VOX (DD * HH * WW)       // 16384
#define NT (VOX / 16)            // 1024 tiles of 16 voxels
#define TPB 256
#define WAVES_PER_BLOCK 8
#define MASKED (-3.0e38f)

// ---------------------------------------------------------------------------
// Kernel 0: zero the reduction scratch (d2max[4] + flags[8])
// ---------------------------------------------------------------------------
__global__ void hd_init(unsigned* d2max, unsigned* flags) {
    int t = threadIdx.x;
    if (t < BATCH) d2max[t] = 0u;
    if (t < 2 * BATCH) flags[t] = 0u;
}

// ---------------------------------------------------------------------------
// Kernel 1: per-voxel masked squared norms + per-batch validity flags.
// sigmoid(x) > 0.5  <=>  x > 0.
// ---------------------------------------------------------------------------
__global__ __launch_bounds__(TPB) void hd_prep(const float* __restrict__ logits,
                                               const float* __restrict__ targets,
                                               float* __restrict__ psq,
                                               float* __restrict__ tsq,
                                               unsigned* __restrict__ flags) {
    int idx = blockIdx.x * TPB + threadIdx.x;   // over BATCH*VOX (exact multiple)
    int b = idx >> 14;                          // / VOX
    int v = idx & (VOX - 1);
    int d = v >> 9;                             // v / (HH*WW)
    int h = (v >> 4) & 31;
    int w = v & 15;
    float sq = (float)(d * d + h * h + w * w);
    bool pm = logits[idx] > 0.0f;
    bool tm = targets[idx] > 0.5f;
    psq[idx] = pm ? sq : MASKED;
    tsq[idx] = tm ? sq : MASKED;
    unsigned long long bp = __ballot(pm);
    unsigned long long bt = __ballot(tm);
    if ((threadIdx.x & 31) == 0) {              // one atomic per wave32
        if (bp) atomicOr(&flags[2 * b + 0], 1u);
        if (bt) atomicOr(&flags[2 * b + 1], 1u);
    }
}

// ---------------------------------------------------------------------------
// Kernel 2: tiled masked max of d2 via V_WMMA_F32_16X16X4_F32, homogeneous
// coordinates so the whole bias lives inside the matmul:
//   A row_i = (d, h, w, 1)              (pred voxel coords)
//   B col_j = (-2d', -2h', -2w', tsq_j) (target coords scaled, masked norm)
//   C       = psq_i broadcast           (LOOP-INVARIANT)
//   D       = psq_i + tsq_j - 2 c_i.c_j = fully-masked d2 tile
// grid = (NT / WAVES_PER_BLOCK, BATCH), block = 256 (8 waves of 32).
// Each wave owns one i-tile (16 pred voxels), loops over all 1024 j-tiles.
// ---------------------------------------------------------------------------
__global__ __launch_bounds__(TPB) void hd_main(const float* __restrict__ psq,
                                               const float* __restrict__ tsq,
                                               unsigned* __restrict__ d2max) {
    __shared__ float lds_tsq[VOX];              // 64 KB of the WGP's 320 KB LDS

    const int b = blockIdx.y;
    const float* tsqb = tsq + b * VOX;

    // Stage this batch's tsq into LDS (b128 loads, ds stores).
    for (int t = threadIdx.x; t < VOX / 4; t += TPB) {
        ((float4*)lds_tsq)[t] = ((const float4*)tsqb)[t];
    }
    __syncthreads();

    const int lane = threadIdx.x & 31;
    const int wave = threadIdx.x >> 5;
    const bool hiHalf = lane >= 16;
    const int n = lane & 15;

    // ---- A operand: homogeneous coords of pred voxels i0..i0+15 ---------
    // 16x4 f32 A layout: lanes 0-15 hold (K0,K1)=(d,h); lanes 16-31 (K2,K3)=(w,1)
    const int i0 = (blockIdx.x * WAVES_PER_BLOCK + wave) * 16;
    const int vi = i0 + n;
    const float di = (float)(vi >> 9);
    const float hi = (float)((vi >> 4) & 31);
    const float wi = (float)(vi & 15);
    v2f amat;
    amat.x = hiHalf ? wi : di;
    amat.y = hiHalf ? 1.0f : hi;

    // ---- loop-invariant C: masked row norms (C/D row M = r + 8*hiHalf) --
    const float* psqb = psq + b * VOX + i0 + (hiHalf ? 8 : 0);
    v8f ps;
#pragma unroll
    for (int r = 0; r < 8; ++r) ps[r] = psqb[r];

    // B K2 row = -2 * w-coordinate = -2 * lane (loop invariant: tile width
    // equals WW, so the w coord of column n is n for every j-tile).
    const float neg2n = -2.0f * (float)n;

    v8f acc;
#pragma unroll
    for (int r = 0; r < 8; ++r) acc[r] = MASKED;

#pragma unroll 4
    for (int jt = 0; jt < NT; ++jt) {
        // j-tile coords: d = jt>>5, h = jt&31 are wave-uniform scalars.
        const float dj = (float)(jt >> 5);
        const float hj = (float)(jt & 31);
        const float ts = lds_tsq[jt * 16 + n];  // masked col norm (ds_load_b32)

        // 4x16 f32 B layout: VGPR0 low/high = rows K0/K2, VGPR1 low/high = K1/K3.
        v2f bmat;
        bmat.x = hiHalf ? neg2n : (-2.0f * dj);
        bmat.y = hiHalf ? ts : (-2.0f * hj);

        // D = A x B + C  ==  psq_i + tsq_j - 2 * c_i . c_j  (masked d2)
        v8f dmat = __builtin_amdgcn_wmma_f32_16x16x4_f32(
            /*neg_a=*/false, amat, /*neg_b=*/false, bmat,
            /*c_mod=*/(short)0, ps, /*reuse_a=*/false, /*reuse_b=*/false);

#pragma unroll
        for (int r = 0; r < 8; ++r) acc[r] = fmaxf(acc[r], dmat[r]);
    }

    // ---- reduce: 8 VGPRs -> lane max -> wave32 max -> per-batch atomic ----
    float m = acc[0];
#pragma unroll
    for (int r = 1; r < 8; ++r) m = fmaxf(m, acc[r]);
    m = fmaxf(m, 0.0f);   // clamp; validity handled by flags
#pragma unroll
    for (int off = 16; off >= 1; off >>= 1)
        m = fmaxf(m, __shfl_xor(m, off, 32));
    if (lane == 0)
        atomicMax(&d2max[b], __float_as_uint(m));  // bits monotone for m >= 0
}

// ---------------------------------------------------------------------------
// Kernel 3: sqrt + validity + batch mean.
// ---------------------------------------------------------------------------
__global__ void hd_final(const unsigned* __restrict__ d2max,
                         const unsigned* __restrict__ flags,
                         float* __restrict__ out) {
    if (threadIdx.x == 0 && blockIdx.x == 0) {
        float s = 0.0f;
        for (int b = 0; b < BATCH; ++b) {
            float m2 = __uint_as_float(d2max[b]);
            bool valid = (flags[2 * b + 0] != 0u) && (flags[2 * b + 1] != 0u);
            s += valid ? sqrtf(m2) : __builtin_inff();
        }
        out[0] = s * (1.0f / BATCH);
    }
}

// ---------------------------------------------------------------------------
extern "C" void kernel_launch(void* const* d_in, const int* in_sizes, int n_in,
                              void* d_out, int out_size, void* d_ws, size_t ws_size,
                              hipStream_t stream) {
    (void)in_sizes; (void)n_in; (void)out_size; (void)ws_size;
    const float* logits  = (const float*)d_in[0];
    const float* targets = (const float*)d_in[1];

    float* psq      = (float*)d_ws;                 // BATCH*VOX floats (256 KB)
    float* tsq      = psq + BATCH * VOX;            // BATCH*VOX floats (256 KB)
    unsigned* d2max = (unsigned*)(tsq + BATCH * VOX);
    unsigned* flags = d2max + BATCH;

    hd_init<<<1, 32, 0, stream>>>(d2max, flags);
    hd_prep<<<(BATCH * VOX) / TPB, TPB, 0, stream>>>(logits, targets, psq, tsq, flags);
    dim3 grid(NT / WAVES_PER_BLOCK, BATCH);
    hd_main<<<grid, TPB, 0, stream>>>(psq, tsq, d2max);
    hd_final<<<1, 32, 0, stream>>>(d2max, flags, (float*)d_out);
}